// PropagateFlow_26920855011397
// MI455X (gfx1250) — compile-verified
//
#include <hip/hip_runtime.h>
#include <math.h>

// ---------------------------------------------------------------------------
// Planar normalizing flow (PropagateFlow), T=16, DIM=4e6, reformulated as:
//   G = W * U^T  (16x16xK GEMM, K=DIM)  -> v_wmma_f32_16x16x4_f32
//   wz0 = W * z0 (fused into same pass)
//   tiny 16-step scalar recurrence -> h[16], logdet
//   z_out = z0 + U^T h  (streaming rank-16 update)
// Bandwidth-bound: ~0.82 GB total traffic -> ~35us at 23.3 TB/s.
// ---------------------------------------------------------------------------

typedef float v2f __attribute__((ext_vector_type(2)));
typedef float v8f __attribute__((ext_vector_type(8)));

#define NT 16                 // number of transforms
#define WAVES_PER_BLOCK 8
#define BLOCK_SZ (WAVES_PER_BLOCK * 32)

// ws layout (floats): [0..255]=G (16x16, row t = w_t, col s = u_s),
//                     [256..271]=wz0, [272..287]=h, [288]=logdet scratch
#define WS_G    0
#define WS_WZ0  256
#define WS_H    272

__global__ void pf_init_ws(float* ws) {
    int i = threadIdx.x;
    if (i < 512) ws[i] = 0.0f;
}

// ---------------------------------------------------------------------------
// Kernel 1: G[t][s] = sum_k w[t][k]*u[s][k]  and  wz0[t] = sum_k w[t][k]*z[k]
// Each wave owns a contiguous K-slice; one v_wmma_f32_16x16x4_f32 per 4 K.
// A-frag layout (16x4 f32): lane l<16 -> row l, K={k0,k0+1};
//                           lane l>=16 -> row l-16, K={k0+2,k0+3}.
// B-frag (4x16 f32, B = U^T so B[k][n] = u[n][k]): same per-lane addressing
// pattern on u.  C (16x16 f32, 8 VGPRs): vgpr j, lanes0-15 -> (M=j, N=lane),
// lanes16-31 -> (M=j+8, N=lane-16).
// Loop bounds are forced into SGPRs (readfirstlane) so the compiler emits a
// scalar-controlled loop: EXEC stays all-1s across the WMMAs, and loop
// control costs SALU instead of per-lane VALU+exec masking.
// ---------------------------------------------------------------------------
__global__ void __launch_bounds__(BLOCK_SZ)
pf_gram_wmma(const float* __restrict__ z, const float* __restrict__ u,
             const float* __restrict__ w, float* __restrict__ ws, int dim)
{
    __shared__ float sG[256];
    __shared__ float sWz[NT];
    const int tid = threadIdx.x;
    if (tid < 256) sG[tid] = 0.0f;
    if (tid < NT) sWz[tid] = 0.0f;
    __syncthreads();

    const int lane = tid & 31;
    const int wave = tid >> 5;
    const int row  = lane & 15;   // A row (t) / B col (s)
    const int hi   = lane >> 4;   // which K-half of the 4-wide chunk

    const long rbase  = (long)row * (long)dim;
    const int nchunk  = dim >> 2;                       // dim % 4 == 0
    const int nwaves  = (int)gridDim.x * WAVES_PER_BLOCK;
    // Wave index is uniform across the wave: pin it to an SGPR so all loop
    // bounds below are scalar (s_cmp/s_cbranch loop, full-EXEC WMMA body).
    const int wid = __builtin_amdgcn_readfirstlane(
                        (int)blockIdx.x * WAVES_PER_BLOCK + wave);
    const int cpw = (nchunk + nwaves - 1) / nwaves;
    const int c0  = wid * cpw;
    const int c1  = (c0 + cpw < nchunk) ? (c0 + cpw) : nchunk;

    v8f acc = {};         // C accumulator: G tile partial
    float wz = 0.0f;      // per-lane partial of w_row . z  (this lane's K's)

    const long koff = (long)(hi << 1);
    int c = c0;

    // Main body: 4 WMMAs (16 K) per trip; 12 b64 loads issued up front for
    // memory-level parallelism, then prefetch ~1KB ahead on the row streams.
    for (; c + 4 <= c1; c += 4) {
        const long k = ((long)c << 2) + koff;
        const float* wp = w + rbase + k;
        const float* up = u + rbase + k;
        const float* zp = z + k;
        v2f a0 = *(const v2f*)(wp);      v2f b0 = *(const v2f*)(up);
        v2f a1 = *(const v2f*)(wp + 4);  v2f b1 = *(const v2f*)(up + 4);
        v2f a2 = *(const v2f*)(wp + 8);  v2f b2 = *(const v2f*)(up + 8);
        v2f a3 = *(const v2f*)(wp + 12); v2f b3 = *(const v2f*)(up + 12);
        v2f z0 = *(const v2f*)(zp);      v2f z1 = *(const v2f*)(zp + 4);
        v2f z2 = *(const v2f*)(zp + 8);  v2f z3 = *(const v2f*)(zp + 12);
        __builtin_prefetch(wp + 256, 0, 0);   // global_prefetch_b8
        __builtin_prefetch(up + 256, 0, 0);

        wz = fmaf(a0.x, z0.x, wz); wz = fmaf(a0.y, z0.y, wz);
        wz = fmaf(a1.x, z1.x, wz); wz = fmaf(a1.y, z1.y, wz);
        wz = fmaf(a2.x, z2.x, wz); wz = fmaf(a2.y, z2.y, wz);
        wz = fmaf(a3.x, z3.x, wz); wz = fmaf(a3.y, z3.y, wz);

        // D = A*B + C; accumulator chaining (D->C) is pipelined on the XDL.
        acc = __builtin_amdgcn_wmma_f32_16x16x4_f32(false, a0, false, b0,
                                                    (short)0, acc, false, false);
        acc = __builtin_amdgcn_wmma_f32_16x16x4_f32(false, a1, false, b1,
                                                    (short)0, acc, false, false);
        acc = __builtin_amdgcn_wmma_f32_16x16x4_f32(false, a2, false, b2,
                                                    (short)0, acc, false, false);
        acc = __builtin_amdgcn_wmma_f32_16x16x4_f32(false, a3, false, b3,
                                                    (short)0, acc, false, false);
    }
    // Remainder (0..3 chunks).
    for (; c < c1; ++c) {
        const long k = ((long)c << 2) + koff;
        v2f a  = *(const v2f*)(w + rbase + k);
        v2f bb = *(const v2f*)(u + rbase + k);
        v2f zz = *(const v2f*)(z + k);
        wz = fmaf(a.x, zz.x, wz);
        wz = fmaf(a.y, zz.y, wz);
        acc = __builtin_amdgcn_wmma_f32_16x16x4_f32(false, a, false, bb,
                                                    (short)0, acc, false, false);
    }

    // wz0: combine the two K-halves (lane l + lane l+16), then LDS-reduce.
    wz += __shfl_xor(wz, 16, 32);
    if (hi == 0) atomicAdd(&sWz[row], wz);

    // G tile: LDS-reduce across the 8 waves of this block.
    #pragma unroll
    for (int j = 0; j < 8; ++j) {
        const int m = j + (hi << 3);
        atomicAdd(&sG[m * 16 + row], acc[j]);
    }
    __syncthreads();

    // One set of global f32 atomics per block (272 values, low contention).
    if (tid < 256) atomicAdd(&ws[WS_G + tid], sG[tid]);
    if (tid < NT)  atomicAdd(&ws[WS_WZ0 + tid], sWz[tid]);
}

// ---------------------------------------------------------------------------
// Kernel 2: the 16-step recurrence (tiny, single thread).
//   inner_t = wz0[t] + b[t] + sum_{s<t} G[t][s] * h_s
//   h_t = tanh(inner_t);  logdet += log|1 + (1-h_t^2)*G[t][t]|
// ---------------------------------------------------------------------------
__global__ void pf_scan(const float* __restrict__ b, float* __restrict__ ws,
                        float* __restrict__ out_logdet)
{
    if (threadIdx.x != 0 || blockIdx.x != 0) return;
    const float* G   = ws + WS_G;
    const float* wz0 = ws + WS_WZ0;
    float* h         = ws + WS_H;
    float hh[NT];
    float ld = 0.0f;
    for (int t = 0; t < NT; ++t) {
        float inner = wz0[t] + b[t];
        for (int s = 0; s < t; ++s)
            inner = fmaf(G[t * 16 + s], hh[s], inner);
        const float ht = tanhf(inner);
        hh[t] = ht;
        h[t]  = ht;
        const float udp = (1.0f - ht * ht) * G[t * 16 + t];
        ld += logf(fabsf(1.0f + udp));
    }
    *out_logdet = ld;
}

// ---------------------------------------------------------------------------
// Kernel 3: z_out = z0 + sum_t h_t * u_t   (streaming, float4 vectorized)
// ---------------------------------------------------------------------------
__global__ void __launch_bounds__(256)
pf_combine(const float* __restrict__ z, const float* __restrict__ u,
           const float* __restrict__ ws, float* __restrict__ out, int dim)
{
    __shared__ float sh[NT];
    if (threadIdx.x < NT) sh[threadIdx.x] = ws[WS_H + threadIdx.x];
    __syncthreads();

    const int n4     = dim >> 2;   // dim % 4 == 0
    const int stride = (int)gridDim.x * (int)blockDim.x;
    for (int i4 = (int)blockIdx.x * (int)blockDim.x + (int)threadIdx.x;
         i4 < n4; i4 += stride) {
        const long i = (long)i4 << 2;
        float4 zz = *(const float4*)(z + i);
        #pragma unroll
        for (int t = 0; t < NT; ++t) {
            const float4 uu = *(const float4*)(u + (long)t * (long)dim + i);
            const float ht = sh[t];
            zz.x = fmaf(uu.x, ht, zz.x);
            zz.y = fmaf(uu.y, ht, zz.y);
            zz.z = fmaf(uu.z, ht, zz.z);
            zz.w = fmaf(uu.w, ht, zz.w);
        }
        *(float4*)(out + i) = zz;
    }
}

// ---------------------------------------------------------------------------
extern "C" void kernel_launch(void* const* d_in, const int* in_sizes, int n_in,
                              void* d_out, int out_size, void* d_ws, size_t ws_size,
                              hipStream_t stream)
{
    const float* z = (const float*)d_in[0];   // [DIM]
    const float* u = (const float*)d_in[1];   // [T, DIM]
    const float* w = (const float*)d_in[2];   // [T, DIM]
    const float* b = (const float*)d_in[3];   // [T, 1]
    float* out = (float*)d_out;               // [DIM] z_out, then [1] logdet
    float* ws  = (float*)d_ws;
    const int dim = in_sizes[0];

    pf_init_ws<<<1, 512, 0, stream>>>(ws);

    const int gram_blocks = 512;              // 4096 waves sharing DIM/4 chunks
    pf_gram_wmma<<<gram_blocks, BLOCK_SZ, 0, stream>>>(z, u, w, ws, dim);

    pf_scan<<<1, 32, 0, stream>>>(b, ws, out + dim);

    const int n4 = dim >> 2;
    int cgrid = (n4 + 255) / 256;
    if (cgrid > 8192) cgrid = 8192;
    pf_combine<<<cgrid, 256, 0, stream>>>(z, u, ws, out, dim);
}